// RTDetrHungarianMatcher_34720515620960
// MI455X (gfx1250) — compile-verified
//
#include <hip/hip_runtime.h>

// RT-DETR Hungarian matcher cost matrix for gfx1250 (MI455X).
//
// cost[i][j] = 5*L1(pred_i, tgt_j) + 2*focal_class(i, label_j) - 2*GIoU(i, j)
//
// Design (VALU-throughput-bound; store floor 123MB @ 23.3TB/s ~ 5us):
//  1) class_cost_kernel: focal cost table cfull[9600][80] with C_CLASS=2
//     pre-folded (exact power-of-two scale). 40x fewer transcendentals than
//     naive per-pair evaluation; table is L2-resident (3 MB).
//  2) cost_kernel: workgroup owns a 16-row query tile (MTILE=16 keeps the
//     kernel under 256 VGPRs -- MTILE=32 forced s_set_vgpr_msb juggling and
//     ~3x static SALU). Per 16-col target tile the class gather
//     cfull[i][labels[j]] runs on the matrix pipe as a one-hot matmul:
//     20 x V_WMMA_F32_16X16X4_F32 (numerically exact). One-hot B is built
//     with immediate-operand compares (q = lbl - 2*half vs literal 4s/4s+1),
//     so no constant table in VGPRs. Box L1 + GIoU run on the VALU and
//     co-execute with the WMMAs. Boxes pre-scaled by C_BBOX=5 (L1 absorbs
//     it; GIoU is scale-invariant). Divides = v_rcp_f32 + mul.

#define NQ 9600          // B*Q queries
#define NT 3200          // B*T targets
#define NC 80            // classes
#define NKSTEP (NC / 4)  // 20 WMMA k-steps of K=4
#define JTILES (NT / 16) // 200 column tiles
#define MTILE 16         // query rows per workgroup

typedef __attribute__((ext_vector_type(2))) float v2f;
typedef __attribute__((ext_vector_type(8))) float v8f;

// ---------------- Kernel 1: focal class-cost table (x C_CLASS) --------------
__global__ __launch_bounds__(256) void class_cost_kernel(
    const float* __restrict__ logits, float* __restrict__ cfull) {
  const int idx = blockIdx.x * 256 + threadIdx.x;  // NQ*NC = 768000, exact grid
  const float x = logits[idx];
  const float p = 1.0f / (1.0f + __expf(-x));
  const float omp = 1.0f - p;
  // GAMMA=2 -> squares; ALPHA=0.25; C_CLASS=2 folded: 2*0.75=1.5, 2*0.25=0.5
  const float neg = 1.5f * p * p * (-__logf(omp + 1e-8f));
  const float pos = 0.5f * omp * omp * (-__logf(p + 1e-8f));
  cfull[idx] = pos - neg;
}

// ---------------- Kernel 2: fused cost matrix -------------------------------
__global__ __launch_bounds__(256) void cost_kernel(
    const float* __restrict__ cfull,
    const float* __restrict__ pred_boxes,   // [NQ][4] cx,cy,w,h
    const int*   __restrict__ labels,       // [NT]
    const float* __restrict__ tgt_boxes,    // [NT][4] cx,cy,w,h
    float* __restrict__ out) {              // [NQ][NT]
  // Per-query data (scaled by C_BBOX=5) staged once in LDS; all inner-loop
  // reads are half-wave broadcasts (conflict-free).
  __shared__ float s_pred[MTILE][9];  // cx,cy,w,h, x0,y0,x1,y1, area

  const int i0 = blockIdx.x * MTILE;

  if (threadIdx.x < MTILE) {
    const float4 pb = ((const float4*)pred_boxes)[i0 + threadIdx.x];
    const float cx = 5.0f * pb.x, cy = 5.0f * pb.y;
    const float w  = 5.0f * pb.z, h  = 5.0f * pb.w;
    const float x0 = cx - 0.5f * w, y0 = cy - 0.5f * h;
    const float x1 = cx + 0.5f * w, y1 = cy + 0.5f * h;
    s_pred[threadIdx.x][0] = cx;
    s_pred[threadIdx.x][1] = cy;
    s_pred[threadIdx.x][2] = w;
    s_pred[threadIdx.x][3] = h;
    s_pred[threadIdx.x][4] = x0;
    s_pred[threadIdx.x][5] = y0;
    s_pred[threadIdx.x][6] = x1;
    s_pred[threadIdx.x][7] = y1;
    s_pred[threadIdx.x][8] = (x1 - x0) * (y1 - y0);
  }
  __syncthreads();

  const int lane = threadIdx.x & 31;
  const int wave = threadIdx.x >> 5;
  const int ln   = lane & 15;  // column within tile / A-fragment row
  const int half = lane >> 4;  // lane half: K pair select (A/B), M+8 (C/D)

  // A fragments in registers for all j-tiles (40 VGPRs).
  // ISA 32-bit 16x4 A layout: VGPR0 -> K = base + 2*half, VGPR1 -> +1.
  v2f a[NKSTEP];
  {
    const float* ar = cfull + (size_t)(i0 + ln) * NC + 2 * half;
#pragma unroll
    for (int s = 0; s < NKSTEP; ++s) {
      a[s].x = ar[4 * s];
      a[s].y = ar[4 * s + 1];
    }
  }

  // Wave w owns j-tiles w, w+8, ... (200/8 = 25 each; uniform trip count ->
  // EXEC stays all-ones as WMMA requires).
  for (int jt = wave; jt < JTILES; jt += 8) {
    const int jg = jt * 16 + ln;
    const float4 tb = ((const float4*)tgt_boxes)[jg];
    // One-hot compare key: fold the lane-half K offset into the label once,
    // so every per-step compare uses an immediate constant (no VGPR table).
    const int q = labels[jg] - 2 * half;
    const float tcx = 5.0f * tb.x, tcy = 5.0f * tb.y;
    const float tw  = 5.0f * tb.z, th  = 5.0f * tb.w;
    const float tx0 = tcx - 0.5f * tw, ty0 = tcy - 0.5f * th;
    const float tx1 = tcx + 0.5f * tw, ty1 = tcy + 0.5f * th;
    const float tarea = (tx1 - tx0) * (ty1 - ty0);

    // One-hot gather as matmul on the matrix pipe.
    v8f c = {};
#pragma unroll
    for (int s = 0; s < NKSTEP; ++s) {
      v2f b;
      b.x = (q == 4 * s)     ? 1.0f : 0.0f;
      b.y = (q == 4 * s + 1) ? 1.0f : 0.0f;
      c = __builtin_amdgcn_wmma_f32_16x16x4_f32(false, a[s], false, b,
                                                (short)0, c, false, false);
    }

    // C/D layout: VGPR r holds (M = r + 8*half, N = ln).
#pragma unroll
    for (int r = 0; r < 8; ++r) {
      const int mrow = r + 8 * half;
      const float pcx = s_pred[mrow][0], pcy = s_pred[mrow][1];
      const float pw  = s_pred[mrow][2], ph  = s_pred[mrow][3];
      const float px0 = s_pred[mrow][4], py0 = s_pred[mrow][5];
      const float px1 = s_pred[mrow][6], py1 = s_pred[mrow][7];
      const float parea = s_pred[mrow][8];

      // 5*L1 (boxes pre-scaled)
      const float bbox = fabsf(pcx - tcx) + fabsf(pcy - tcy) +
                         fabsf(pw - tw) + fabsf(ph - th);

      // GIoU (scale-invariant under the x5 pre-scale)
      const float ilx = fmaxf(px0, tx0), ily = fmaxf(py0, ty0);
      const float irx = fminf(px1, tx1), iry = fminf(py1, ty1);
      const float iw = fmaxf(irx - ilx, 0.0f), ih = fmaxf(iry - ily, 0.0f);
      const float inter = iw * ih;
      const float uni = parea + tarea - inter;
      const float elx = fminf(px0, tx0), ely = fminf(py0, ty0);
      const float erx = fmaxf(px1, tx1), ery = fmaxf(py1, ty1);
      const float ew = fmaxf(erx - elx, 0.0f), eh = fmaxf(ery - ely, 0.0f);
      const float enc = ew * eh;
      // giou = inter/uni - (enc-uni)/enc = inter*rcp(uni) + uni*rcp(enc) - 1
      const float giou = inter * __builtin_amdgcn_rcpf(uni) +
                         uni * __builtin_amdgcn_rcpf(enc) - 1.0f;

      // c[r] already carries C_CLASS
      out[(size_t)(i0 + mrow) * NT + jg] = fmaf(-2.0f, giou, bbox + c[r]);
    }
  }
}

// ---------------- Launch ----------------------------------------------------
extern "C" void kernel_launch(void* const* d_in, const int* in_sizes, int n_in,
                              void* d_out, int out_size, void* d_ws, size_t ws_size,
                              hipStream_t stream) {
  const float* logits = (const float*)d_in[0];   // [32,300,80]
  const float* pred   = (const float*)d_in[1];   // [32,300,4]
  const int*   labels = (const int*)d_in[2];     // [3200]
  const float* tboxes = (const float*)d_in[3];   // [3200,4]
  float* out   = (float*)d_out;                  // [32,300,3200] f32
  float* cfull = (float*)d_ws;                   // 9600*80 f32 = 3.07 MB scratch

  class_cost_kernel<<<(NQ * NC) / 256, 256, 0, stream>>>(logits, cfull);
  cost_kernel<<<NQ / MTILE, 256, 0, stream>>>(cfull, pred, labels, tboxes, out);
}